// AdaptiveWaveletFreqModule_25580825215627
// MI455X (gfx1250) — compile-verified
//
#include <hip/hip_runtime.h>
#include <hip/hip_bf16.h>

typedef __attribute__((ext_vector_type(2))) float v2f;
typedef __attribute__((ext_vector_type(8))) float v8f;

#define B_  8
#define C_  64
#define H_  256
#define W_  256
#define OH_ 131
#define OW_ 131
#define NP_ (OH_*OW_)    // 17161 spatial positions per subband map
#define HW_ (H_*W_)      // 65536

// db4 decomposition low-pass filter (pywt order)
__constant__ float DEC_LO[8] = {
  -0.010597401784997278f,  0.032883011666982945f,  0.030841381835986965f,
  -0.18703481171888114f,  -0.02798376941698385f,   0.6308807679295904f,
   0.7148465705525415f,    0.23037781330885523f };

__device__ __forceinline__ int symref(int i, int n) {
  if (i < 0)  i = -i - 1;          // symmetric (edge-inclusive) reflection, p=6 < n
  if (i >= n) i = 2*n - 1 - i;
  return i;
}

// D(16x16,f32) = A(16x4,f32) x B(4x16,f32) + C   — V_WMMA_F32_16X16X4_F32
__device__ __forceinline__ v8f wmma4(v2f a, v2f b, v8f c) {
  return __builtin_amdgcn_wmma_f32_16x16x4_f32(false, a, false, b, (short)0, c,
                                               false, false);
}

// ---------------------------------------------------------------------------
// Kernel 1: DWT2 (db4, symmetric). x(B,C,H,W) -> ff(B, 4C, OH, OW), ch = s*C+c
// Subband order [aa, da, ad, dd]; g_lo[j]=DEC_LO[7-j], g_hi[j]=(-1)^j DEC_LO[j]
// ---------------------------------------------------------------------------
__global__ void dwt2_kernel(const float* __restrict__ x, float* __restrict__ ff) {
  size_t idx = (size_t)blockIdx.x * blockDim.x + threadIdx.x;
  const size_t total = (size_t)B_ * C_ * OH_ * OW_;
  if (idx >= total) return;
  int ow = (int)(idx % OW_);
  int t  = (int)(idx / OW_);
  int oh = t % OH_;  t /= OH_;
  int c  = t % C_;
  int b  = t / C_;

  float glo[8], ghi[8];
  #pragma unroll
  for (int j = 0; j < 8; ++j) {
    glo[j] = DEC_LO[7 - j];
    ghi[j] = (j & 1) ? -DEC_LO[j] : DEC_LO[j];
  }
  int rows[8], cols[8];
  #pragma unroll
  for (int i = 0; i < 8; ++i) {
    rows[i] = symref(2*oh + i - 6, H_);
    cols[i] = symref(2*ow + i - 6, W_);
  }
  const float* xp = x + (size_t)(b * C_ + c) * HW_;
  float aa = 0.f, da = 0.f, ad = 0.f, dd = 0.f;
  #pragma unroll
  for (int i = 0; i < 8; ++i) {
    const float* rp = xp + (size_t)rows[i] * W_;
    float rlo = 0.f, rhi = 0.f;
    #pragma unroll
    for (int j = 0; j < 8; ++j) {
      float v = rp[cols[j]];
      rlo += glo[j] * v;
      rhi += ghi[j] * v;
    }
    aa += glo[i] * rlo;  da += ghi[i] * rlo;   // da: hi on rows (vertical)
    ad += glo[i] * rhi;  dd += ghi[i] * rhi;
  }
  size_t base = ((size_t)b * (4*C_) + c) * NP_ + (size_t)oh * OW_ + ow;
  const size_t cs = (size_t)C_ * NP_;
  ff[base]        = aa;
  ff[base + cs]   = da;
  ff[base + 2*cs] = ad;
  ff[base + 3*cs] = dd;
}

// ---------------------------------------------------------------------------
// Kernel 2: z(128 x NP) = w1(128x256) x ff_b(256 x NP) + b1  via WMMA f32
// block = 128 thr = 4 waves; wave w -> M-tiles [32w,32w+16),[32w+16,32w+32);
// block -> 64 N cols. Each B fragment feeds 2 WMMAs (1.25 vmem per WMMA).
// ---------------------------------------------------------------------------
__global__ void gemm1_wmma(const float* __restrict__ ff, const float* __restrict__ w1,
                           const float* __restrict__ b1, float* __restrict__ z) {
  const int Kd = 256;
  int b    = blockIdx.y;
  int wave = threadIdx.x >> 5, lane = threadIdx.x & 31;
  int half = lane >> 4,        l16  = lane & 15;
  int m0   = wave * 32;
  int n0   = blockIdx.x * 64;
  const float* ffb = ff + (size_t)b * Kd * NP_;

  int col[4], colc[4];
  #pragma unroll
  for (int s = 0; s < 4; ++s) {
    col[s]  = n0 + s*16 + l16;
    colc[s] = col[s] < NP_ ? col[s] : NP_ - 1;  // clamp loads; stores predicated
  }
  v8f acc0[4] = {}, acc1[4] = {};
  for (int k0 = 0; k0 < Kd; k0 += 4) {
    const float* ap0 = w1 + (size_t)(m0 + l16) * Kd + (k0 + 2*half);
    const float* ap1 = ap0 + (size_t)16 * Kd;
    v2f a0; a0.x = ap0[0]; a0.y = ap0[1];             // A[m][k0+2h / +1]
    v2f a1; a1.x = ap1[0]; a1.y = ap1[1];
    const float* bp = ffb + (size_t)(k0 + 2*half) * NP_;
    #pragma unroll
    for (int s = 0; s < 4; ++s) {
      v2f bb; bb.x = bp[colc[s]]; bb.y = bp[NP_ + colc[s]];  // B[k][n], B[k+1][n]
      acc0[s] = wmma4(a0, bb, acc0[s]);
      acc1[s] = wmma4(a1, bb, acc1[s]);
    }
  }
  float* zb = z + (size_t)b * 128 * NP_;
  #pragma unroll
  for (int r = 0; r < 8; ++r) {
    int mA = m0 + r + 8*half;                // C/D layout: VGPR r -> rows r, r+8
    int mB = mA + 16;
    float biasA = b1[mA], biasB = b1[mB];
    #pragma unroll
    for (int s = 0; s < 4; ++s)
      if (col[s] < NP_) {
        zb[(size_t)mA * NP_ + col[s]] = acc0[s][r] + biasA;
        zb[(size_t)mB * NP_ + col[s]] = acc1[s][r] + biasB;
      }
  }
}

// ---------------------------------------------------------------------------
// Kernel 3: GroupNorm stats. one block per (b,g); 16 ch x NP contiguous floats
// ---------------------------------------------------------------------------
__global__ void gn_stats(const float* __restrict__ z, float* __restrict__ mu,
                         float* __restrict__ rs) {
  int bg = blockIdx.x;                       // 0..63 == b*8+g
  const size_t n = (size_t)16 * NP_;
  const float* p = z + (size_t)bg * n;
  float s = 0.f, ss = 0.f;
  for (size_t i = threadIdx.x; i < n; i += 256) {
    float v = p[i]; s += v; ss += v * v;
  }
  __shared__ float sh[256], sh2[256];
  sh[threadIdx.x] = s; sh2[threadIdx.x] = ss;
  __syncthreads();
  for (int o = 128; o > 0; o >>= 1) {
    if ((int)threadIdx.x < o) { sh[threadIdx.x] += sh[threadIdx.x+o]; sh2[threadIdx.x] += sh2[threadIdx.x+o]; }
    __syncthreads();
  }
  if (threadIdx.x == 0) {
    float m = sh[0] / (float)n;
    float v = sh2[0] / (float)n - m * m;     // biased var (ddof=0)
    mu[bg] = m;
    rs[bg] = rsqrtf(v + 1e-5f);
  }
}

// ---------------------------------------------------------------------------
// Kernel 4: z2 = gelu_exact(GN(z))   (z2 aliases the dead ff buffer)
// ---------------------------------------------------------------------------
__global__ void normgelu_kernel(const float* __restrict__ z, const float* __restrict__ mu,
                                const float* __restrict__ rs, const float* __restrict__ gamma,
                                const float* __restrict__ beta, float* __restrict__ z2) {
  size_t idx = (size_t)blockIdx.x * blockDim.x + threadIdx.x;
  const size_t total = (size_t)B_ * 128 * NP_;
  if (idx >= total) return;
  int m  = (int)((idx / NP_) % 128);
  int bb = (int)(idx / ((size_t)128 * NP_));
  int g  = m >> 4;                            // 16 channels per group
  float v = (z[idx] - mu[bb*8 + g]) * rs[bb*8 + g] * gamma[m] + beta[m];
  z2[idx] = 0.5f * v * (1.0f + erff(v * 0.7071067811865475f));
}

// ---------------------------------------------------------------------------
// Kernel 5a/5b: density (per b,c nonzero fraction) -> tiny MLP -> sigmoid gates
// ---------------------------------------------------------------------------
__global__ void density_kernel(const float* __restrict__ x, float* __restrict__ dens) {
  int bc = blockIdx.x;                       // 0..511
  const float* p = x + (size_t)bc * HW_;
  float cnt = 0.f;
  for (int i = threadIdx.x; i < HW_; i += 256) cnt += (p[i] != 0.0f) ? 1.0f : 0.0f;
  __shared__ float sh[256];
  sh[threadIdx.x] = cnt; __syncthreads();
  for (int o = 128; o > 0; o >>= 1) {
    if ((int)threadIdx.x < o) sh[threadIdx.x] += sh[threadIdx.x+o];
    __syncthreads();
  }
  if (threadIdx.x == 0) dens[bc] = sh[0] * (1.0f / (float)HW_);
}

__global__ void gates_kernel(const float* __restrict__ dens, const float* __restrict__ g1,
                             const float* __restrict__ gb1, const float* __restrict__ g2,
                             const float* __restrict__ gb2, float* __restrict__ gates) {
  int b = threadIdx.x;
  if (b >= B_) return;
  float h[16];
  for (int j = 0; j < 16; ++j) {
    float s = gb1[j];
    for (int c = 0; c < 64; ++c) s += dens[b*64 + c] * g1[j*64 + c];
    h[j] = s > 0.f ? s : 0.f;
  }
  for (int s4 = 0; s4 < 4; ++s4) {
    float s = gb2[s4];
    for (int j = 0; j < 16; ++j) s += h[j] * g2[s4*16 + j];
    gates[b*4 + s4] = 1.0f / (1.0f + expf(-s));
  }
}

// ---------------------------------------------------------------------------
// Kernel 6: gated(256 x NP) = gate[b, m/64] * (w2(256x128) x z2_b + b2)  WMMA
// block = 256 thr = 8 waves; wave w -> M-tiles 32w, 32w+16 (covers 256 rows).
// A 16-row tile never crosses a subband boundary (m0 multiple of 16).
// ---------------------------------------------------------------------------
__global__ void gemm2_wmma(const float* __restrict__ z2, const float* __restrict__ w2,
                           const float* __restrict__ b2, const float* __restrict__ gates,
                           float* __restrict__ gated) {
  const int Kd = 128;
  int b    = blockIdx.y;
  int wave = threadIdx.x >> 5, lane = threadIdx.x & 31;
  int half = lane >> 4,        l16  = lane & 15;
  int m0   = wave * 32;
  int n0   = blockIdx.x * 64;
  const float* zb = z2 + (size_t)b * Kd * NP_;

  int col[4], colc[4];
  #pragma unroll
  for (int s = 0; s < 4; ++s) {
    col[s]  = n0 + s*16 + l16;
    colc[s] = col[s] < NP_ ? col[s] : NP_ - 1;
  }
  v8f acc0[4] = {}, acc1[4] = {};
  for (int k0 = 0; k0 < Kd; k0 += 4) {
    const float* ap0 = w2 + (size_t)(m0 + l16) * Kd + (k0 + 2*half);
    const float* ap1 = ap0 + (size_t)16 * Kd;
    v2f a0; a0.x = ap0[0]; a0.y = ap0[1];
    v2f a1; a1.x = ap1[0]; a1.y = ap1[1];
    const float* bp = zb + (size_t)(k0 + 2*half) * NP_;
    #pragma unroll
    for (int s = 0; s < 4; ++s) {
      v2f bb; bb.x = bp[colc[s]]; bb.y = bp[NP_ + colc[s]];
      acc0[s] = wmma4(a0, bb, acc0[s]);
      acc1[s] = wmma4(a1, bb, acc1[s]);
    }
  }
  float gateA = gates[b*4 + (m0 >> 6)];           // subband of rows m0..m0+15
  float gateB = gates[b*4 + ((m0 + 16) >> 6)];    // subband of rows m0+16..m0+31
  float* gb = gated + (size_t)b * 256 * NP_;
  #pragma unroll
  for (int r = 0; r < 8; ++r) {
    int mA = m0 + r + 8*half;
    int mB = mA + 16;
    float biasA = b2[mA], biasB = b2[mB];
    #pragma unroll
    for (int s = 0; s < 4; ++s)
      if (col[s] < NP_) {
        gb[(size_t)mA * NP_ + col[s]] = (acc0[s][r] + biasA) * gateA;
        gb[(size_t)mB * NP_ + col[s]] = (acc1[s][r] + biasB) * gateB;
      }
  }
}

// ---------------------------------------------------------------------------
// Kernel 7: IDWT2 (transposed conv, lhs_dilation=2, pad 1). 4x4 taps/subband.
// Effective taps: K[s][i][j] = fa[i]*fb[j], lo tap = DEC_LO[i],
// hi tap = (-1)^(i+1) DEC_LO[7-i]  (== dec_hi[i])
// ---------------------------------------------------------------------------
__global__ void idwt2_kernel(const float* __restrict__ gated, float* __restrict__ recon) {
  size_t idx = (size_t)blockIdx.x * blockDim.x + threadIdx.x;
  const size_t total = (size_t)B_ * C_ * HW_;
  if (idx >= total) return;
  int w = (int)(idx % W_);
  int t = (int)(idx / W_);
  int h = t % H_;  t /= H_;
  int c = t % C_;
  int b = t / C_;

  float flo[8], fhi[8];
  #pragma unroll
  for (int i = 0; i < 8; ++i) {
    flo[i] = DEC_LO[i];
    fhi[i] = (i & 1) ? DEC_LO[7 - i] : -DEC_LO[7 - i];
  }
  const float* gb = gated + (size_t)b * (4*C_) * NP_;
  const size_t cs = (size_t)C_ * NP_;
  int i0 = (h + 1) & 1, j0 = (w + 1) & 1;   // parity so (h+i-1) is even
  float acc = 0.f;
  #pragma unroll
  for (int ii = 0; ii < 4; ++ii) {
    int i = i0 + 2*ii, num = h + i - 1;
    if (num < 0) continue;
    int m = num >> 1; if (m >= OH_) continue;
    #pragma unroll
    for (int jj = 0; jj < 4; ++jj) {
      int j = j0 + 2*jj, nw = w + j - 1;
      if (nw < 0) continue;
      int n = nw >> 1; if (n >= OW_) continue;
      size_t off = (size_t)c * NP_ + (size_t)m * OW_ + n;
      float gaa = gb[off], gda = gb[off + cs], gad = gb[off + 2*cs], gdd = gb[off + 3*cs];
      acc += flo[i]*flo[j]*gaa + fhi[i]*flo[j]*gda + flo[i]*fhi[j]*gad + fhi[i]*fhi[j]*gdd;
    }
  }
  recon[idx] = acc;
}

// ---------------------------------------------------------------------------
// Kernel 8: out(64 x HW) = wf(64x128) x concat[x; recon]_b + bf   via WMMA
// block = 128 thr = 4 waves: 2 M-pairs x 2 N-halves; block covers 128 N cols.
// ---------------------------------------------------------------------------
__global__ void fused_conv_wmma(const float* __restrict__ x, const float* __restrict__ recon,
                                const float* __restrict__ wf, const float* __restrict__ bf,
                                float* __restrict__ out) {
  const int Kd = 128;
  int b    = blockIdx.y;
  int wave = threadIdx.x >> 5, lane = threadIdx.x & 31;
  int half = lane >> 4,        l16  = lane & 15;
  int m0   = (wave & 1) * 32;                       // tiles {0,1} or {2,3}
  int n0   = blockIdx.x * 128 + (wave >> 1) * 64;
  const float* xb = x     + (size_t)b * C_ * HW_;
  const float* rb = recon + (size_t)b * C_ * HW_;

  v8f acc0[4] = {}, acc1[4] = {};
  for (int k0 = 0; k0 < Kd; k0 += 4) {
    const float* ap0 = wf + (size_t)(m0 + l16) * Kd + (k0 + 2*half);
    const float* ap1 = ap0 + (size_t)16 * Kd;
    v2f a0; a0.x = ap0[0]; a0.y = ap0[1];
    v2f a1; a1.x = ap1[0]; a1.y = ap1[1];
    const float* bsrc = (k0 < 64) ? (xb + (size_t)(k0      + 2*half) * HW_)
                                  : (rb + (size_t)(k0 - 64 + 2*half) * HW_);
    #pragma unroll
    for (int s = 0; s < 4; ++s) {
      int cc = n0 + s*16 + l16;
      v2f bb; bb.x = bsrc[cc]; bb.y = bsrc[HW_ + cc];
      acc0[s] = wmma4(a0, bb, acc0[s]);
      acc1[s] = wmma4(a1, bb, acc1[s]);
    }
  }
  float* ob = out + (size_t)b * C_ * HW_;
  #pragma unroll
  for (int r = 0; r < 8; ++r) {
    int mA = m0 + r + 8*half;
    int mB = mA + 16;
    float biasA = bf[mA], biasB = bf[mB];
    #pragma unroll
    for (int s = 0; s < 4; ++s) {
      size_t cc = (size_t)(n0 + s*16 + l16);
      ob[(size_t)mA * HW_ + cc] = acc0[s][r] + biasA;
      ob[(size_t)mB * HW_ + cc] = acc1[s][r] + biasB;
    }
  }
}

// ---------------------------------------------------------------------------
extern "C" void kernel_launch(void* const* d_in, const int* in_sizes, int n_in,
                              void* d_out, int out_size, void* d_ws, size_t ws_size,
                              hipStream_t stream) {
  (void)in_sizes; (void)n_in; (void)out_size; (void)ws_size;
  const float* x     = (const float*)d_in[0];
  const float* w1    = (const float*)d_in[1];
  const float* b1    = (const float*)d_in[2];
  const float* gamma = (const float*)d_in[3];
  const float* beta  = (const float*)d_in[4];
  const float* w2    = (const float*)d_in[5];
  const float* b2    = (const float*)d_in[6];
  const float* g1w   = (const float*)d_in[7];
  const float* gb1   = (const float*)d_in[8];
  const float* g2w   = (const float*)d_in[9];
  const float* gb2   = (const float*)d_in[10];
  const float* wf    = (const float*)d_in[11];
  const float* bf    = (const float*)d_in[12];
  float* out = (float*)d_out;
  float* ws  = (float*)d_ws;

  // workspace layout (floats): ff | z | gated | recon | stats  (~486 MB total)
  const size_t FF_N = (size_t)B_ * 256 * NP_;   // 35,145,728
  const size_t Z_N  = (size_t)B_ * 128 * NP_;   // 17,572,864
  const size_t G_N  = FF_N;
  const size_t R_N  = (size_t)B_ * C_ * HW_;    // 33,554,432
  float* ff    = ws;
  float* z     = ff + FF_N;
  float* gated = z + Z_N;
  float* recon = gated + G_N;
  float* mu    = recon + R_N;
  float* rs    = mu + 64;
  float* dens  = rs + 64;
  float* gts   = dens + 512;
  float* z2    = ff;                            // ff is dead after GEMM1

  {
    size_t tot = (size_t)B_ * C_ * OH_ * OW_;
    dwt2_kernel<<<(unsigned)((tot + 255) / 256), 256, 0, stream>>>(x, ff);
  }
  dim3 grid_g1((NP_ + 63) / 64, B_);
  gemm1_wmma<<<grid_g1, 128, 0, stream>>>(ff, w1, b1, z);
  gn_stats<<<64, 256, 0, stream>>>(z, mu, rs);
  {
    size_t tot = Z_N;
    normgelu_kernel<<<(unsigned)((tot + 255) / 256), 256, 0, stream>>>(z, mu, rs, gamma, beta, z2);
  }
  density_kernel<<<512, 256, 0, stream>>>(x, dens);
  gates_kernel<<<1, 32, 0, stream>>>(dens, g1w, gb1, g2w, gb2, gts);
  dim3 grid_g2((NP_ + 63) / 64, B_);
  gemm2_wmma<<<grid_g2, 256, 0, stream>>>(z2, w2, b2, gts, gated);
  {
    size_t tot = R_N;
    idwt2_kernel<<<(unsigned)((tot + 255) / 256), 256, 0, stream>>>(gated, recon);
  }
  dim3 grid_f(HW_ / 128, B_);
  fused_conv_wmma<<<grid_f, 128, 0, stream>>>(x, recon, wf, bf, out);
}